// KGAT_89146341196446
// MI455X (gfx1250) — compile-verified
//
#include <hip/hip_runtime.h>
#include <hip/hip_bf16.h>

typedef float v2f __attribute__((ext_vector_type(2)));
typedef float v8f __attribute__((ext_vector_type(8)));

#define N_USERS_C 30000

// ---------------------------------------------------------------------------
// zero fill
// ---------------------------------------------------------------------------
__global__ void kgat_zero_kernel(float* __restrict__ p, int n) {
    int i = blockIdx.x * blockDim.x + threadIdx.x;
    if (i < n) p[i] = 0.0f;
}

// ---------------------------------------------------------------------------
// scatter: h_neigh[row] += feats[col] * val   (64 threads per edge, 1 dim each)
// feats is split across two tables at `split` (layer1: user/entity; layer2:
// split==N so always table A).
// ---------------------------------------------------------------------------
__global__ void kgat_scatter_kernel(const int* __restrict__ erow,
                                    const int* __restrict__ ecol,
                                    const float* __restrict__ evals,
                                    const float* __restrict__ featsA,
                                    const float* __restrict__ featsB,
                                    int split,
                                    float* __restrict__ hn,
                                    int nE) {
    int gid = blockIdx.x * blockDim.x + threadIdx.x;
    int e = gid >> 6;          // edge index
    int d = gid & 63;          // feature dim
    if (e >= nE) return;
    int row = erow[e];
    int col = ecol[e];
    float v = evals[e];
    const float* frow = (col < split) ? (featsA + (size_t)col * 64)
                                      : (featsB + (size_t)(col - split) * 64);
    atomicAdd(&hn[(size_t)row * 64 + d], frow[d] * v);
}

// ---------------------------------------------------------------------------
// WMMA GNN layer:  h = leaky_relu((f+hn)@W1 + b1 + (f*hn)@W2 + b2); L2-normalize
// DIN = 64 fixed. One wave32 per 16-row tile; 4 waves / block.
// A layout (16x4 f32): lane(0-15)=row M, VGPR0/1 = K,K+1; lanes 16-31 hold K+2,K+3.
// B layout mirrored: lane(0-15)=col N, VGPR0/1 = K,K+1 (halves offset by 2).
// C/D layout: VGPR r -> row M = r + 8*(lane/16), col N = lane%16.
// ---------------------------------------------------------------------------
template <int DOUT>
__global__ void __launch_bounds__(128)
kgat_gnn_layer_kernel(const float* __restrict__ featsA,
                      const float* __restrict__ featsB,
                      int split,
                      const float* __restrict__ hn,
                      const float* __restrict__ W1, const float* __restrict__ b1,
                      const float* __restrict__ W2, const float* __restrict__ b2,
                      float* __restrict__ out, int n_nodes) {
    constexpr int DIN = 64;
    constexpr int NTILES = DOUT / 16;

    __shared__ float sW1[DIN * DOUT];
    __shared__ float sW2[DIN * DOUT];
    __shared__ float sB[DOUT];

    for (int i = threadIdx.x; i < DIN * DOUT; i += blockDim.x) {
        sW1[i] = W1[i];
        sW2[i] = W2[i];
    }
    if (threadIdx.x < DOUT) sB[threadIdx.x] = b1[threadIdx.x] + b2[threadIdx.x];
    __syncthreads();

    const int wave = threadIdx.x >> 5;
    const int lane = threadIdx.x & 31;
    const int tile = blockIdx.x * 4 + wave;      // 16-row tile index
    const int n_tiles = n_nodes >> 4;            // n_nodes is a multiple of 16
    if (tile >= n_tiles) return;                 // wave-uniform -> EXEC stays all-1s

    const int lc = lane & 15;                    // row (A) / col (B,C) within tile
    const int hi = lane >> 4;                    // half-wave selector

    const int m = (tile << 4) + lc;
    const float* frow = (m < split) ? (featsA + (size_t)m * DIN)
                                    : (featsB + (size_t)(m - split) * DIN);
    const float* hrow = hn + (size_t)m * DIN;

    // Preload A operands for all 16 K-steps (K=4 each): X = f + h, Y = f * h.
    v2f ax[16], ay[16];
#pragma unroll
    for (int ki = 0; ki < 16; ++ki) {
        int k = 4 * ki + 2 * hi;
        float2 f = *(const float2*)(frow + k);
        float2 h = *(const float2*)(hrow + k);
        v2f xa; xa[0] = f.x + h.x; xa[1] = f.y + h.y;
        v2f ya; ya[0] = f.x * h.x; ya[1] = f.y * h.y;
        ax[ki] = xa;
        ay[ki] = ya;
    }

    v8f acc[NTILES];
#pragma unroll
    for (int nt = 0; nt < NTILES; ++nt) {
        v8f c = {};
        const int n = nt * 16 + lc;
#pragma unroll
        for (int ki = 0; ki < 16; ++ki) {
            int k = 4 * ki + 2 * hi;
            v2f bw1; bw1[0] = sW1[k * DOUT + n]; bw1[1] = sW1[(k + 1) * DOUT + n];
            c = __builtin_amdgcn_wmma_f32_16x16x4_f32(false, ax[ki], false, bw1,
                                                      (short)0, c, false, false);
            v2f bw2; bw2[0] = sW2[k * DOUT + n]; bw2[1] = sW2[(k + 1) * DOUT + n];
            c = __builtin_amdgcn_wmma_f32_16x16x4_f32(false, ay[ki], false, bw2,
                                                      (short)0, c, false, false);
        }
        acc[nt] = c;
    }

    // bias + leaky relu
#pragma unroll
    for (int nt = 0; nt < NTILES; ++nt) {
        float bn = sB[nt * 16 + lc];
#pragma unroll
        for (int r = 0; r < 8; ++r) {
            float x = acc[nt][r] + bn;
            acc[nt][r] = (x > 0.0f) ? x : 0.01f * x;
        }
    }

    // per-row L2 norm: row (r + 8*hi) lives on the 16 lanes of this half-wave
    float inv[8];
#pragma unroll
    for (int r = 0; r < 8; ++r) {
        float s = 0.0f;
#pragma unroll
        for (int nt = 0; nt < NTILES; ++nt) s += acc[nt][r] * acc[nt][r];
        s += __shfl_xor(s, 8, 16);
        s += __shfl_xor(s, 4, 16);
        s += __shfl_xor(s, 2, 16);
        s += __shfl_xor(s, 1, 16);
        float nrm = sqrtf(s);
        inv[r] = 1.0f / fmaxf(nrm, 1e-12f);
    }

    // store normalized rows
#pragma unroll
    for (int nt = 0; nt < NTILES; ++nt) {
#pragma unroll
        for (int r = 0; r < 8; ++r) {
            int row = (tile << 4) + r + 8 * hi;
            out[(size_t)row * DOUT + nt * 16 + lc] = acc[nt][r] * inv[r];
        }
    }
}

// ---------------------------------------------------------------------------
// scoring: one wave32 per pair, 160-dim dot across a0 (split tables), a1, a2
// ---------------------------------------------------------------------------
__global__ void kgat_score_kernel(const int* __restrict__ uids,
                                  const int* __restrict__ iids,
                                  const float* __restrict__ ue,
                                  const float* __restrict__ ee,
                                  const float* __restrict__ a1,
                                  const float* __restrict__ a2,
                                  float* __restrict__ out,
                                  int nB, int nUsers) {
    int wave = threadIdx.x >> 5;
    int lane = threadIdx.x & 31;
    int b = blockIdx.x * (blockDim.x >> 5) + wave;
    if (b >= nB) return;
    int u = uids[b];
    int it = iids[b];

    const float* u0 = ue + (size_t)u * 64;
    const float* i0 = ee + (size_t)it * 64;             // a0 entity part
    const float* u1 = a1 + (size_t)u * 64;
    const float* i1 = a1 + (size_t)(nUsers + it) * 64;
    const float* u2 = a2 + (size_t)u * 32;
    const float* i2 = a2 + (size_t)(nUsers + it) * 32;

    float s = u0[lane] * i0[lane] + u0[lane + 32] * i0[lane + 32]
            + u1[lane] * i1[lane] + u1[lane + 32] * i1[lane + 32]
            + u2[lane] * i2[lane];

    s += __shfl_xor(s, 16, 32);
    s += __shfl_xor(s, 8, 32);
    s += __shfl_xor(s, 4, 32);
    s += __shfl_xor(s, 2, 32);
    s += __shfl_xor(s, 1, 32);
    if (lane == 0) out[b] = s;
}

// ---------------------------------------------------------------------------
extern "C" void kernel_launch(void* const* d_in, const int* in_sizes, int n_in,
                              void* d_out, int out_size, void* d_ws, size_t ws_size,
                              hipStream_t stream) {
    const int*   erow  = (const int*)d_in[0];
    const int*   ecol  = (const int*)d_in[1];
    const float* evals = (const float*)d_in[2];
    const float* ue    = (const float*)d_in[3];
    const float* ee    = (const float*)d_in[4];
    const float* W1_0  = (const float*)d_in[5];
    const float* b1_0  = (const float*)d_in[6];
    const float* W2_0  = (const float*)d_in[7];
    const float* b2_0  = (const float*)d_in[8];
    const float* W1_1  = (const float*)d_in[9];
    const float* b1_1  = (const float*)d_in[10];
    const float* W2_1  = (const float*)d_in[11];
    const float* b2_1  = (const float*)d_in[12];
    const int*   uids  = (const int*)d_in[13];
    const int*   iids  = (const int*)d_in[14];

    const int E_  = in_sizes[0];
    const int NU  = in_sizes[3] / 64;   // 30000
    const int NE_ = in_sizes[4] / 64;   // 70000
    const int N_  = NU + NE_;           // 100000 (multiple of 16)
    const int B_  = out_size;           // 4096

    float* ws = (float*)d_ws;
    float* a1 = ws;                              // N * 64
    float* a2 = ws + (size_t)N_ * 64;            // N * 32
    float* hn = ws + (size_t)N_ * 96;            // N * 64

    const int nHN = N_ * 64;
    const int zeroBlocks = (nHN + 255) / 256;
    const long long nScatter = (long long)E_ * 64;
    const int scatterBlocks = (int)((nScatter + 255) / 256);
    const int gemmBlocks = (N_ / 16 + 3) / 4;
    const int scoreBlocks = (B_ + 7) / 8;

    // ---- layer 1 ----
    kgat_zero_kernel<<<zeroBlocks, 256, 0, stream>>>(hn, nHN);
    kgat_scatter_kernel<<<scatterBlocks, 256, 0, stream>>>(
        erow, ecol, evals, ue, ee, NU, hn, E_);
    kgat_gnn_layer_kernel<64><<<gemmBlocks, 128, 0, stream>>>(
        ue, ee, NU, hn, W1_0, b1_0, W2_0, b2_0, a1, N_);

    // ---- layer 2 ----
    kgat_zero_kernel<<<zeroBlocks, 256, 0, stream>>>(hn, nHN);
    kgat_scatter_kernel<<<scatterBlocks, 256, 0, stream>>>(
        erow, ecol, evals, a1, a1, N_, hn, E_);
    kgat_gnn_layer_kernel<32><<<gemmBlocks, 128, 0, stream>>>(
        a1, a1, N_, hn, W1_1, b1_1, W2_1, b2_1, a2, N_);

    // ---- scoring ----
    kgat_score_kernel<<<scoreBlocks, 256, 0, stream>>>(
        uids, iids, ue, ee, a1, a2, (float*)d_out, B_, NU);
}